// HDPHMM_41721312314403
// MI455X (gfx1250) — compile-verified
//
#include <hip/hip_runtime.h>
#include <hip/hip_bf16.h>

typedef __attribute__((ext_vector_type(2))) float v2f;
typedef __attribute__((ext_vector_type(8))) float v8f;

// Problem constants
#define TT 262144
#define KK 20
#define FF 16
#define CHUNKS 1024
#define LL 256            // TT / CHUNKS

// Workspace layout (float offsets)
#define OFF_P    0        // 400  P row-major
#define OFF_PC   512      // 400  P column-major: Pc[j*20+l] = P[l][j]
#define OFF_W    1024     // 32   stick-breaking weights (padded)
#define OFF_CST  1056     // 32   per-state constant (padded)
#define OFF_B    1088     // 1024 GEMM B matrix [32][32] (k padded to 32)
#define OFF_SB   4096     // CHUNKS*20 forward boundary alphas
#define OFF_DB   24576    // CHUNKS*20 backward boundary betas
#define OFF_MC   45056    // CHUNKS*400 chunk transfer matrices
#define OFF_E    454656   // TT*20 E = exp(em)

#ifndef __has_builtin
#define __has_builtin(x) 0
#endif

#if defined(__HIP_DEVICE_COMPILE__) && \
    __has_builtin(__builtin_amdgcn_tensor_load_to_lds) && \
    __has_builtin(__builtin_amdgcn_s_wait_tensorcnt)
#define USE_TDM 1
#else
#define USE_TDM 0
#endif

#if USE_TDM
typedef __attribute__((ext_vector_type(4))) unsigned int u32x4;
typedef __attribute__((ext_vector_type(4))) int i32x4;
typedef __attribute__((ext_vector_type(8))) int i32x8;

// Issue one TDM load of a 1 x tile_elems f32 tile; reads beyond tensor_elems
// are clamped to zero by the tensor unit (OOB rule), replacing manual guards.
// D# bit layout per CDNA5 ISA ch.8: group0 = {flags, lds_addr, global_addr,
// type=2}; group1 = {data_size=4B, tensor_dim0, tensor_dim1=1, tile_dim0,
// tile_dim1=1, stride}; groups 2/3 set dims 2..4 = 1.
__device__ __forceinline__ void tdm_load_1d(void* lds_dst, const void* gsrc,
                                            unsigned tile_elems,
                                            unsigned tensor_elems) {
  unsigned long long ga = (unsigned long long)(__SIZE_TYPE__)gsrc;
  unsigned lds = (unsigned)(__SIZE_TYPE__)lds_dst;  // low 32 bits = LDS offset
  u32x4 g0 = {1u, lds, (unsigned)ga,
              ((unsigned)((ga >> 32) & 0x01FFFFFFull)) | 0x80000000u};
  i32x8 g1 = {(int)(2u << 16),                       // data_size = 4 bytes
              (int)(tensor_elems << 16),             // tensor_dim0[15:0]
              (int)((tensor_elems >> 16) | (1u << 16)), // dim0 hi | tensor_dim1=1
              (int)(tile_elems << 16),               // tile_dim0
              1,                                     // tile_dim1 = 1
              (int)tile_elems,                       // tensor_dim0_stride
              0, 0};
  i32x4 g2 = {1, 1, 0, 0};                           // tensor_dim2/3 = 1
  i32x4 g3 = {0, (int)(1u << 16), 0, 0};             // tensor_dim4 = 1
#if defined(__clang_major__) && (__clang_major__ >= 23)
  i32x8 g4 = {0, 0, 0, 0, 0, 0, 0, 0};
  __builtin_amdgcn_tensor_load_to_lds(g0, g1, g2, g3, g4, 0);
#else
  __builtin_amdgcn_tensor_load_to_lds(g0, g1, g2, g3, 0);
#endif
}
#endif

// Stage nfloats contiguous f32 (of which `avail` are readable) into LDS.
__device__ __forceinline__ void load_floats_lds(float* sdst, const float* gsrc,
                                                int nfloats, long avail,
                                                int lane) {
#if USE_TDM
  (void)lane;  // TDM issues once per wave; EXEC ignored
  unsigned te = (avail < (long)nfloats) ? (unsigned)avail : (unsigned)nfloats;
  tdm_load_1d(sdst, gsrc, (unsigned)nfloats, te);
#else
  const float4* g4p = (const float4*)gsrc;
  float4* s4 = (float4*)sdst;
  for (int i = lane; i < nfloats / 4; i += 32)
    if ((long)i * 4 + 3 < avail) s4[i] = g4p[i];
#endif
}

__device__ __forceinline__ void lds_load_fence() {
#if USE_TDM
  __builtin_amdgcn_s_wait_tensorcnt(0);
#endif
}

// ---------------------------------------------------------------------------
// Preprocessing: stick-breaking w, softmax P (+transpose), GEMM operands.
// ---------------------------------------------------------------------------
__global__ __launch_bounds__(32) void k_pre(const float* __restrict__ bl,
                                            const float* __restrict__ pi,
                                            const float* __restrict__ means,
                                            const float* __restrict__ lv,
                                            float* __restrict__ ws) {
  const int lane = threadIdx.x;
  if (lane == 0) {
    float cp = 1.f;
    for (int k = 0; k < KK; ++k) {
      float b = 1.f / (1.f + __expf(-bl[k]));
      ws[OFF_W + k] = b * cp;
      cp *= (1.f - b);
    }
    for (int k = KK; k < 32; ++k) ws[OFF_W + k] = 0.f;
  }
  if (lane < KK) {
    float mx = -3.4e38f;
    for (int j = 0; j < KK; ++j) mx = fmaxf(mx, pi[lane * KK + j]);
    float e[KK]; float s = 0.f;
    for (int j = 0; j < KK; ++j) { e[j] = __expf(pi[lane * KK + j] - mx); s += e[j]; }
    float is = 1.f / s;
    for (int j = 0; j < KK; ++j) {
      float p = e[j] * is;
      ws[OFF_P + lane * KK + j] = p;
      ws[OFF_PC + j * KK + lane] = p;
    }
    float cst = 0.f;
    for (int f = 0; f < FF; ++f) {
      float var = __expf(lv[lane * FF + f]) + 1e-6f;
      float iv = 1.f / var;
      float mn = means[lane * FF + f];
      ws[OFF_B + f * 32 + lane]        = iv;
      ws[OFF_B + (16 + f) * 32 + lane] = -2.f * mn * iv;
      cst += mn * mn * iv + __logf(6.283185307179586f * var);
    }
    ws[OFF_CST + lane] = cst;
  } else {
    for (int f = 0; f < 32; ++f) ws[OFF_B + f * 32 + lane] = 0.f;
    ws[OFF_CST + lane] = 0.f;
  }
}

// ---------------------------------------------------------------------------
// Emission: E[t][k] = exp(-0.5*(A.B + cst[k])) via V_WMMA_F32_16X16X4_F32.
// A[t][r] = r<16 ? obs[t][r]^2 : obs[t][r-16];  B[r][k] = inv / -2*mean*inv.
// Block = 256 threads (8 waves), each wave owns 16 T-rows, 2 k-tiles.
// ---------------------------------------------------------------------------
__global__ __launch_bounds__(256) void k_em(const float* __restrict__ obs,
                                            float* __restrict__ ws) {
  __shared__ float sObs[128 * 16];
  __shared__ float sB[32 * 32];
  __shared__ float sCst[32];
  const int tid = threadIdx.x;
  const int t0 = blockIdx.x * 128;

  {
    const float4* g4 = (const float4*)(obs + (long)t0 * FF);
    float4* s4 = (float4*)sObs;
    for (int i = tid; i < 512; i += 256) s4[i] = g4[i];
    const float4* b4 = (const float4*)(ws + OFF_B);
    float4* sb4 = (float4*)sB;
    if (tid < 256) sb4[tid] = b4[tid];
    if (tid < 32) sCst[tid] = ws[OFF_CST + tid];
  }
  __syncthreads();

  const int lane = tid & 31;
  const int wave = tid >> 5;
  const int m = lane & 15;
  const int half = lane >> 4;
  const int rowBase = wave * 16;

  v8f acc0 = {};
  v8f acc1 = {};
#pragma unroll
  for (int s = 0; s < 8; ++s) {
    const int r0 = 4 * s + 2 * half;
    const int r1 = r0 + 1;
    v2f a, b0, b1;
    {
      float o0 = sObs[(rowBase + m) * 16 + (r0 & 15)];
      float o1 = sObs[(rowBase + m) * 16 + (r1 & 15)];
      a.x = (r0 < 16) ? o0 * o0 : o0;
      a.y = (r1 < 16) ? o1 * o1 : o1;
    }
    b0.x = sB[r0 * 32 + m];       b0.y = sB[r1 * 32 + m];
    b1.x = sB[r0 * 32 + 16 + m];  b1.y = sB[r1 * 32 + 16 + m];
    acc0 = __builtin_amdgcn_wmma_f32_16x16x4_f32(false, a, false, b0, (short)0, acc0, false, false);
    acc1 = __builtin_amdgcn_wmma_f32_16x16x4_f32(false, a, false, b1, (short)0, acc1, false, false);
  }

  float* E = ws + OFF_E;
#pragma unroll
  for (int v = 0; v < 8; ++v) {
    const int row = rowBase + v + 8 * half;
    const int t = t0 + row;
    {
      const int k = m;
      float em = -0.5f * (acc0[v] + sCst[k]);
      E[t * KK + k] = __expf(em);
    }
    {
      const int k = 16 + m;
      if (k < KK) {
        float em = -0.5f * (acc1[v] + sCst[k]);
        E[t * KK + k] = __expf(em);
      }
    }
  }
}

// ---------------------------------------------------------------------------
// Phase 1: chunk transfer matrices  M_c = prod_{t=cL+1}^{min((c+1)L,T-1)} P*diag(E[t])
// One wave per chunk; lane i owns matrix row i; per-step scalar renorm.
// E window + P staged into LDS by the Tensor Data Mover.
// ---------------------------------------------------------------------------
__global__ __launch_bounds__(32) void k_chain(float* __restrict__ ws) {
  __shared__ float sPc[32 * KK];
  __shared__ float sE[(LL + 1) * KK];
  const int lane = threadIdx.x;
  const int c = blockIdx.x;
  const long base = (long)c * LL;

  load_floats_lds(sPc, ws + OFF_PC, 400, 400, lane);
  load_floats_lds(sE, ws + OFF_E + base * KK, (LL + 1) * KK,
                  ((long)TT - base) * KK, lane);
  lds_load_fence();
  __syncthreads();

  float mrow[KK];
#pragma unroll
  for (int j = 0; j < KK; ++j) mrow[j] = (lane == j) ? 1.f : 0.f;

  const int tEnd = (c == CHUNKS - 1) ? (LL - 1) : LL;   // last chunk stops at T-1
  for (int tt = 1; tt <= tEnd; ++tt) {
    float acc[KK];
    float rs = 0.f;
#pragma unroll
    for (int j = 0; j < KK; ++j) {
      const float4* pc = (const float4*)&sPc[j * KK];
      float s = 0.f;
#pragma unroll
      for (int q = 0; q < 5; ++q) {
        float4 p = pc[q];
        s = fmaf(mrow[4 * q + 0], p.x, s);
        s = fmaf(mrow[4 * q + 1], p.y, s);
        s = fmaf(mrow[4 * q + 2], p.z, s);
        s = fmaf(mrow[4 * q + 3], p.w, s);
      }
      acc[j] = s * sE[tt * KK + j];
      rs += acc[j];
    }
    float tot = rs;
#pragma unroll
    for (int off = 16; off; off >>= 1) tot += __shfl_xor(tot, off, 32);
    const float sc = 1.f / (tot + 1e-30f);
#pragma unroll
    for (int j = 0; j < KK; ++j) mrow[j] = acc[j] * sc;
  }

  if (lane < KK) {
    float* Mc = ws + OFF_MC + (long)c * 400 + lane * KK;
#pragma unroll
    for (int j = 0; j < KK; ++j) Mc[j] = mrow[j];
  }
}

// ---------------------------------------------------------------------------
// Phase 2: sequential boundary chains. Block 0: forward alphas. Block 1: backward betas.
// global_prefetch of the next chunk matrix hides the only global latency.
// ---------------------------------------------------------------------------
__global__ __launch_bounds__(32) void k_bound(float* __restrict__ ws) {
  __shared__ float buf[32];
  const int lane = threadIdx.x;

  if (blockIdx.x == 0) {
    float x = (lane < KK) ? ws[OFF_W + lane] * ws[OFF_E + lane] : 0.f;
    buf[lane] = x;
    __syncthreads();
    float s[KK]; float tot = 0.f;
#pragma unroll
    for (int j = 0; j < KK; ++j) { s[j] = buf[j]; tot += s[j]; }
    float sc = 1.f / (tot + 1e-10f);
#pragma unroll
    for (int j = 0; j < KK; ++j) s[j] *= sc;
    if (lane < KK) ws[OFF_SB + lane] = x * sc;

    for (int c = 0; c < CHUNKS - 1; ++c) {
      const float* Mc = ws + OFF_MC + (long)c * 400;
      __builtin_prefetch(Mc + 400 + lane * 16, 0, 3);  // next chunk matrix
      float xx = 0.f;
#pragma unroll
      for (int l = 0; l < KK; ++l) xx = fmaf(s[l], Mc[l * KK + lane], xx);
      __syncthreads();
      buf[lane] = xx;
      __syncthreads();
      tot = 0.f;
#pragma unroll
      for (int j = 0; j < KK; ++j) { s[j] = buf[j]; tot += s[j]; }
      sc = 1.f / (tot + 1e-10f);
#pragma unroll
      for (int j = 0; j < KK; ++j) s[j] *= sc;
      if (lane < KK) ws[OFF_SB + (long)(c + 1) * KK + lane] = xx * sc;
    }
  } else {
    float d[KK];
#pragma unroll
    for (int j = 0; j < KK; ++j) d[j] = 1.f;
    if (lane < KK) ws[OFF_DB + (long)(CHUNKS - 1) * KK + lane] = 1.f;

    for (int c = CHUNKS - 1; c >= 1; --c) {
      const float* McB = ws + OFF_MC + (long)c * 400;
      __builtin_prefetch(McB - 400 + lane * 16, 0, 3); // next (lower) chunk matrix
      const float4* mr = (const float4*)(McB + lane * KK);
      float xx = 0.f;
#pragma unroll
      for (int q = 0; q < 5; ++q) {
        float4 p = mr[q];
        xx = fmaf(d[4 * q + 0], p.x, xx);
        xx = fmaf(d[4 * q + 1], p.y, xx);
        xx = fmaf(d[4 * q + 2], p.z, xx);
        xx = fmaf(d[4 * q + 3], p.w, xx);
      }
      __syncthreads();
      buf[lane] = xx;
      __syncthreads();
      float tot = 0.f;
#pragma unroll
      for (int j = 0; j < KK; ++j) { d[j] = buf[j]; tot += d[j]; }
      const float sc = 1.f / (tot + 1e-10f);
#pragma unroll
      for (int j = 0; j < KK; ++j) d[j] *= sc;
      if (lane < KK) ws[OFF_DB + (long)(c - 1) * KK + lane] = xx * sc;
    }
  }
}

// ---------------------------------------------------------------------------
// Phase 3: parallel replay. blocks [0,C): forward alpha; [C,2C): backward beta.
// ---------------------------------------------------------------------------
__global__ __launch_bounds__(32) void k_replay(float* __restrict__ ws,
                                               float* __restrict__ out) {
  __shared__ float sP[32 * KK];
  __shared__ float sE[(LL + 1) * KK];
  __shared__ float buf[32];
  const int lane = threadIdx.x;
  const int dir = (blockIdx.x >= CHUNKS) ? 1 : 0;
  const int c = blockIdx.x - dir * CHUNKS;
  const long base = (long)c * LL;

  const int pOff = dir ? OFF_P : OFF_PC;
  load_floats_lds(sP, ws + pOff, 400, 400, lane);
  load_floats_lds(sE, ws + OFF_E + base * KK, (LL + 1) * KK,
                  ((long)TT - base) * KK, lane);
  lds_load_fence();
  __syncthreads();

  float* alpha = out;
  float* beta = out + (long)TT * KK;

  if (!dir) {
    float a[KK];
    const float av = (lane < KK) ? ws[OFF_SB + (long)c * KK + lane] : 0.f;
    buf[lane] = av;
    __syncthreads();
#pragma unroll
    for (int j = 0; j < KK; ++j) a[j] = buf[j];
    if (lane < KK) alpha[base * KK + lane] = av;

    for (int tt = 1; tt <= LL - 1; ++tt) {
      const float4* pc = (const float4*)&sP[lane * KK];  // column `lane` of P
      float s = 0.f;
#pragma unroll
      for (int q = 0; q < 5; ++q) {
        float4 p = pc[q];
        s = fmaf(a[4 * q + 0], p.x, s);
        s = fmaf(a[4 * q + 1], p.y, s);
        s = fmaf(a[4 * q + 2], p.z, s);
        s = fmaf(a[4 * q + 3], p.w, s);
      }
      const float x = s * sE[tt * KK + lane];
      __syncthreads();
      buf[lane] = x;
      __syncthreads();
      float tot = 0.f;
#pragma unroll
      for (int j = 0; j < KK; ++j) { a[j] = buf[j]; tot += a[j]; }
      const float sc = 1.f / (tot + 1e-10f);
#pragma unroll
      for (int j = 0; j < KK; ++j) a[j] *= sc;
      if (lane < KK) alpha[(base + tt) * KK + lane] = x * sc;
    }
    if (c == CHUNKS - 1) {
      float s2 = 0.f;
#pragma unroll
      for (int j = 0; j < KK; ++j) s2 += a[j];
      if (lane == 0) out[(long)2 * TT * KK] = __logf(s2 + 1e-10f);
    }
  } else {
    float b[KK];
    int tHi;
    if (c == CHUNKS - 1) {
#pragma unroll
      for (int j = 0; j < KK; ++j) b[j] = 1.f;
      if (lane < KK) beta[(long)(TT - 1) * KK + lane] = 1.f;
      tHi = LL - 2;
    } else {
      const float bv = (lane < KK) ? ws[OFF_DB + (long)c * KK + lane] : 0.f;
      buf[lane] = bv;
      __syncthreads();
#pragma unroll
      for (int j = 0; j < KK; ++j) b[j] = buf[j];
      tHi = LL - 1;
    }

    for (int tt = tHi; tt >= 0; --tt) {
      float y[KK];
#pragma unroll
      for (int j = 0; j < KK; ++j) y[j] = b[j] * sE[(tt + 1) * KK + j];
      const float4* pr = (const float4*)&sP[lane * KK];  // row `lane` of P
      float s = 0.f;
#pragma unroll
      for (int q = 0; q < 5; ++q) {
        float4 p = pr[q];
        s = fmaf(y[4 * q + 0], p.x, s);
        s = fmaf(y[4 * q + 1], p.y, s);
        s = fmaf(y[4 * q + 2], p.z, s);
        s = fmaf(y[4 * q + 3], p.w, s);
      }
      __syncthreads();
      buf[lane] = s;
      __syncthreads();
      float tot = 0.f;
#pragma unroll
      for (int j = 0; j < KK; ++j) { b[j] = buf[j]; tot += b[j]; }
      const float sc = 1.f / (tot + 1e-10f);
#pragma unroll
      for (int j = 0; j < KK; ++j) b[j] *= sc;
      if (lane < KK) beta[(base + tt) * KK + lane] = s * sc;
    }
  }
}

// ---------------------------------------------------------------------------
extern "C" void kernel_launch(void* const* d_in, const int* in_sizes, int n_in,
                              void* d_out, int out_size, void* d_ws, size_t ws_size,
                              hipStream_t stream) {
  const float* obs   = (const float*)d_in[0];
  const float* bl    = (const float*)d_in[1];
  const float* pi    = (const float*)d_in[2];
  const float* means = (const float*)d_in[3];
  const float* lv    = (const float*)d_in[4];
  float* out = (float*)d_out;
  float* ws  = (float*)d_ws;

  k_pre<<<1, 32, 0, stream>>>(bl, pi, means, lv, ws);
  k_em<<<TT / 128, 256, 0, stream>>>(obs, ws);
  k_chain<<<CHUNKS, 32, 0, stream>>>(ws);
  k_bound<<<2, 32, 0, stream>>>(ws);
  k_replay<<<2 * CHUNKS, 32, 0, stream>>>(ws, out);
}